// ScorePredictor_79285096284696
// MI455X (gfx1250) — compile-verified
//
#include <hip/hip_runtime.h>

typedef float v2f __attribute__((ext_vector_type(2)));
typedef float v8f __attribute__((ext_vector_type(8)));

#define NDIM   128   // input feature dim D
#define NCH    128   // output channels C
#define BM     64    // edges per block tile
#define LDS_RS 132   // padded LDS row stride in dwords (132 % 64 == 4 -> conflict-free A frags)

__global__ __launch_bounds__(256)
void score_predictor_wmma(const float* __restrict__ features,
                          const int*   __restrict__ src,
                          const int*   __restrict__ dst,
                          const float* __restrict__ Ww,
                          const float* __restrict__ Wb,
                          const float* __restrict__ Xw,
                          const float* __restrict__ Xb,
                          float* __restrict__ outW,
                          float* __restrict__ outX,
                          int nE)
{
    __shared__ __align__(16) float sS[BM * LDS_RS];

    const int tid   = threadIdx.x;
    const int eBase = blockIdx.x * BM;

    // ---------------- stage score tile: S[e][d] = f[src[e]][d] - f[dst[e]][d]
    {
        const int eL = tid >> 2;           // 0..63 : edge within tile
        const int cg = (tid & 3) * 32;     // 32-float column group
        const int e  = eBase + eL;
        float4* dl = (float4*)&sS[eL * LDS_RS + cg];
        if (e < nE) {
            const long long si = src[e];
            const long long di = dst[e];
            const float4* fs = (const float4*)(features + (size_t)si * NDIM + cg);
            const float4* fd = (const float4*)(features + (size_t)di * NDIM + cg);
            #pragma unroll
            for (int j = 0; j < 8; ++j) {
                float4 a = fs[j];
                float4 b = fd[j];
                float4 r;
                r.x = a.x - b.x; r.y = a.y - b.y;
                r.z = a.z - b.z; r.w = a.w - b.w;
                dl[j] = r;
            }
        } else {
            float4 z; z.x = 0.f; z.y = 0.f; z.z = 0.f; z.w = 0.f;
            #pragma unroll
            for (int j = 0; j < 8; ++j) dl[j] = z;
        }
    }
    __syncthreads();

    // ---------------- per-wave WMMA GEMM: 16 channels x 64 edges, both matrices
    const int lane = tid & 31;
    const int wv   = tid >> 5;     // wave id 0..7 -> channel tile
    const int lm   = lane & 15;    // N (col) / M (row) index within tile
    const int lh   = lane >> 4;    // K-half selector
    const int c0   = wv * 16;

    // B fragment source rows: W[c][k] with c = c0+lm, k pair = 4s + 2*lh + {0,1}
    const float* wrow = Ww + (size_t)(c0 + lm) * NDIM + 2 * lh;
    const float* xrow = Xw + (size_t)(c0 + lm) * NDIM + 2 * lh;

    v8f accW[4];
    v8f accX[4];
    const v8f vzero = {0.f, 0.f, 0.f, 0.f, 0.f, 0.f, 0.f, 0.f};
    #pragma unroll
    for (int m = 0; m < 4; ++m) { accW[m] = vzero; accX[m] = vzero; }

    #pragma unroll 4
    for (int s = 0; s < NDIM / 4; ++s) {
        const int kb = s * 4;
        const v2f bw = *(const v2f*)(wrow + kb);   // W weights fragment (L2-resident)
        const v2f bx = *(const v2f*)(xrow + kb);   // X weights fragment
        #pragma unroll
        for (int m = 0; m < 4; ++m) {
            const v2f a = *(const v2f*)(&sS[(m * 16 + lm) * LDS_RS + kb + 2 * lh]);
            accW[m] = __builtin_amdgcn_wmma_f32_16x16x4_f32(
                false, a, false, bw, (short)0, accW[m], false, false);
            accX[m] = __builtin_amdgcn_wmma_f32_16x16x4_f32(
                false, a, false, bx, (short)0, accX[m], false, false);
        }
    }

    // ---------------- epilogue: bias + store
    const float bW = Wb[c0 + lm];
    const float bX = Xb[c0 + lm];
    #pragma unroll
    for (int m = 0; m < 4; ++m) {
        #pragma unroll
        for (int v = 0; v < 8; ++v) {
            const int e = eBase + m * 16 + 8 * lh + v;   // C/D layout: M = v + 8*lh
            if (e < nE) {
                outW[(size_t)e * NCH + c0 + lm] = accW[m][v] + bW;
                outX[(size_t)e * NCH + c0 + lm] = accX[m][v] + bX;
            }
        }
    }
}

extern "C" void kernel_launch(void* const* d_in, const int* in_sizes, int n_in,
                              void* d_out, int out_size, void* d_ws, size_t ws_size,
                              hipStream_t stream) {
    const float* features = (const float*)d_in[0];
    const int*   src      = (const int*)d_in[1];
    const int*   dst      = (const int*)d_in[2];
    const float* Ww       = (const float*)d_in[3];
    const float* Wb       = (const float*)d_in[4];
    const float* Xw       = (const float*)d_in[5];
    const float* Xb       = (const float*)d_in[6];

    const int nE = in_sizes[1];                 // 640000 edges
    float* outW = (float*)d_out;
    float* outX = outW + (size_t)nE * NCH;      // tuple output: (w, x) concatenated

    const int nBlocks = (nE + BM - 1) / BM;     // 10000
    score_predictor_wmma<<<nBlocks, 256, 0, stream>>>(
        features, src, dst, Ww, Wb, Xw, Xb, outW, outX, nE);
}